// ParallelMamba_12893491822622
// MI455X (gfx1250) — compile-verified
//
#include <hip/hip_runtime.h>
#include <hip/hip_bf16.h>

typedef __attribute__((ext_vector_type(2))) float v2f;
typedef __attribute__((ext_vector_type(8))) float v8f;

#define KC 64  // K-chunk staged per LDS buffer
#define SWS 68 // LDS row stride in floats: 68*4=272B -> 16B aligned, no bank conflicts

// ---------------------------------------------------------------------------
// GEMM: C[m,n] = sum_k A[m,k] * W[n,k]  (+ bias[n])
// A: M x K (row stride lda), W: N x K (row stride ldw), C: M x N (row stride N)
// f32 WMMA 16x16x4. Block = 128 threads = 4 waves stacked along M (64 rows);
// each wave owns a 16 x (16*NT) tile. The W tile (16*NT rows x KC k) is shared
// by all 4 waves: staged into LDS with async global->LDS copies (ASYNCcnt),
// double-buffered so the copy of chunk kc+KC overlaps the WMMAs of chunk kc.
// ---------------------------------------------------------------------------
template <int NT>
__global__ __launch_bounds__(128) void gemm_f32_wmma(
    const float* __restrict__ A, const float* __restrict__ W,
    const float* __restrict__ bias, float* __restrict__ C,
    int M, int N, int K, int lda, int ldw) {
  const int tid  = threadIdx.x;
  const int lane = tid & 31;
  const int wid  = tid >> 5;              // 4 waves per block, stacked along M
  const int half = lane >> 4;             // K-half select (K 0,1 vs 2,3)
  const int r    = lane & 15;             // row (A) / col (B) within 16-tile

  const int m0 = blockIdx.y * 64 + wid * 16;
  const int n0 = blockIdx.x * (16 * NT);

  __shared__ float sW[2 * 16 * NT * SWS];

  // async-stage W[n0 .. n0+16*NT) rows, cols [kc, kc+KC) into LDS buffer `buf`
  auto stage = [&](int buf, int kc) {
#pragma unroll
    for (int it = 0; it < 2 * NT; ++it) {
      const int i   = tid + it * 128;     // 16B transfer index
      const int row = i >> 4;             // 0 .. 16*NT-1
      const int cg  = i & 15;             // 4-float column group
      unsigned lds = (unsigned)(size_t)&sW[buf * (16 * NT * SWS) + row * SWS + cg * 4];
      const float* g = W + (size_t)(n0 + row) * ldw + kc + cg * 4;
      asm volatile("global_load_async_to_lds_b128 %0, %1, off"
                   :: "v"(lds), "v"(g) : "memory");
    }
  };

  v8f acc[NT];
#pragma unroll
  for (int t = 0; t < NT; ++t) acc[t] = (v8f){};

  const float* Aptr = A + (size_t)(m0 + r) * lda + 2 * half;

  stage(0, 0);
  int buf = 0;
  for (int kc = 0; kc < K; kc += KC) {
    asm volatile("s_wait_asynccnt 0x0" ::: "memory");  // this buffer's data landed
    __syncthreads();
    if (kc + KC < K) {
      stage(buf ^ 1, kc + KC);                         // overlap next copy
      __builtin_prefetch(Aptr + kc + KC, 0, 1);        // warm next A chunk in L1
    }
    const float* sw = &sW[buf * (16 * NT * SWS)];
#pragma unroll
    for (int k = 0; k < KC; k += 4) {
      v2f af = *(const v2f*)(Aptr + kc + k);           // global_load_b64, const ofs
#pragma unroll
      for (int t = 0; t < NT; ++t) {
        v2f bf = *(const v2f*)&sw[(16 * t + r) * SWS + 2 * half + k];  // ds_load_b64
        acc[t] = __builtin_amdgcn_wmma_f32_16x16x4_f32(
            false, af, false, bf, (short)0, acc[t], false, false);
      }
    }
    __syncthreads();                                   // all reads of `buf` done
    buf ^= 1;
  }

#pragma unroll
  for (int t = 0; t < NT; ++t) {
    const int col = n0 + 16 * t + r;
    const float bv = bias ? bias[col] : 0.0f;
#pragma unroll
    for (int v = 0; v < 8; ++v) {
      C[(size_t)(m0 + v + 8 * half) * N + col] = acc[t][v] + bv;
    }
  }
}

// ---------------------------------------------------------------------------
// Depthwise causal conv1d (k=3, left pad 2) + SiLU.
// XR: (B*L, 2D) row-major; channel slice [0,D) is xc.  U: (B*L, D).
// ---------------------------------------------------------------------------
__global__ __launch_bounds__(256) void conv_silu_kernel(
    const float* __restrict__ XR, const float* __restrict__ conv_w,
    const float* __restrict__ conv_b, float* __restrict__ U,
    int B, int L, int D, int ldxr) {
  int idx = blockIdx.x * blockDim.x + threadIdx.x;  // (b*L + l)*D + d
  int d = idx % D;
  int l = (idx / D) % L;
  int b = idx / (D * L);
  if (b >= B) return;

  const float w0 = conv_w[d * 3 + 0];
  const float w1 = conv_w[d * 3 + 1];
  const float w2 = conv_w[d * 3 + 2];
  const float* xc = XR + (size_t)(b * L) * ldxr + d;

  float s = conv_b[d] + w2 * xc[(size_t)l * ldxr];
  if (l >= 1) s += w1 * xc[(size_t)(l - 1) * ldxr];
  if (l >= 2) s += w0 * xc[(size_t)(l - 2) * ldxr];
  U[(size_t)idx] = s / (1.0f + __expf(-s));  // SiLU
}

// ---------------------------------------------------------------------------
// Chunked selective scan + output gate.
// One thread per (b, chunk, d); 16-state recurrence in registers.
// Per-chunk B/C (256 steps x 32 floats = 32 KB) staged in LDS, broadcast reads.
// ---------------------------------------------------------------------------
__global__ __launch_bounds__(256) void scan_gate_kernel(
    const float* __restrict__ DR,   // raw delta (T, D), pre-softplus
    const float* __restrict__ U,    // (T, D)
    const float* __restrict__ XP,   // (T, 96): [64..79]=B, [80..95]=C
    const float* __restrict__ XR,   // (T, 2D): res at col offset D
    const float* __restrict__ A_log,// (D, 16)
    float* __restrict__ Y,          // (T, D) gated output
    int L, int D, int chunk, int ldxr) {
  const int d  = blockIdx.x * 256 + threadIdx.x;
  const int c  = blockIdx.y;
  const int b  = blockIdx.z;
  const int t0 = b * L + c * chunk;  // first token of this chunk

  __shared__ float sBC[256 * 32];    // 32 KB
  for (int i = threadIdx.x; i < chunk * 32; i += 256) {
    int t = i >> 5, j = i & 31;
    sBC[i] = XP[(size_t)(t0 + t) * 96 + 64 + j];
  }
  __syncthreads();

  float a[16], carry[16];
#pragma unroll
  for (int n = 0; n < 16; ++n) {
    a[n] = -__expf(A_log[(size_t)d * 16 + n]);
    carry[n] = 0.0f;
  }

  for (int t = 0; t < chunk; ++t) {
    const size_t tok = (size_t)(t0 + t);
    const float dr = DR[tok * D + d];
    const float delta = (dr > 20.0f) ? dr : log1pf(__expf(dr));  // softplus
    const float du = delta * U[tok * D + d];
    const float* bc = &sBC[t * 32];

    float y = 0.0f;
#pragma unroll
    for (int n = 0; n < 16; ++n) {
      const float dA = __expf(delta * a[n]);
      carry[n] = dA * carry[n] + du * bc[n];
      y = fmaf(carry[n], bc[16 + n], y);
    }
    const float res = XR[tok * ldxr + D + d];
    Y[tok * D + d] = y * (res / (1.0f + __expf(-res)));  // * silu(res)
  }
}

// ---------------------------------------------------------------------------
extern "C" void kernel_launch(void* const* d_in, const int* in_sizes, int n_in,
                              void* d_out, int out_size, void* d_ws, size_t ws_size,
                              hipStream_t stream) {
  (void)in_sizes; (void)n_in; (void)out_size; (void)ws_size;
  const float* x      = (const float*)d_in[0];
  const float* W_in   = (const float*)d_in[1];
  const float* b_in   = (const float*)d_in[2];
  const float* conv_w = (const float*)d_in[3];
  const float* conv_b = (const float*)d_in[4];
  const float* W_x    = (const float*)d_in[5];
  const float* W_dt   = (const float*)d_in[6];
  const float* b_dt   = (const float*)d_in[7];
  const float* A_log  = (const float*)d_in[8];
  const float* W_out  = (const float*)d_in[9];
  const float* b_out  = (const float*)d_in[10];
  float* out = (float*)d_out;

  const int B = 2, L = 1024, DM = 1024, D = 2048, R = 64, CH = 256;
  const int T = B * L;  // 2048 tokens

  float* XR = (float*)d_ws;              // (T, 2D)  32 MB
  float* U  = XR + (size_t)T * 2 * D;    // (T, D)   16 MB
  float* XP = U  + (size_t)T * D;        // (T, 96)  0.75 MB
  float* DR = XP + (size_t)T * 96;       // (T, D)   16 MB
  float* Y  = DR + (size_t)T * D;        // (T, D)   16 MB

  dim3 gblk(128);

  // 1) in_proj: XR = x @ W_in^T + b_in          M=2048 N=4096 K=1024
  gemm_f32_wmma<4><<<dim3((2 * D) / 64, T / 64), gblk, 0, stream>>>(
      x, W_in, b_in, XR, T, 2 * D, DM, DM, DM);

  // 2) depthwise causal conv + SiLU -> U
  conv_silu_kernel<<<(B * L * D) / 256, 256, 0, stream>>>(
      XR, conv_w, conv_b, U, B, L, D, 2 * D);

  // 3) x_proj: XP = U @ W_x^T                   M=2048 N=96 K=2048
  gemm_f32_wmma<2><<<dim3(96 / 32, T / 64), gblk, 0, stream>>>(
      U, W_x, nullptr, XP, T, 96, D, D, D);

  // 4) dt proj: DR = dlt @ W_dt^T + b_dt        M=2048 N=2048 K=64 (lda=96)
  gemm_f32_wmma<4><<<dim3(D / 64, T / 64), gblk, 0, stream>>>(
      XP, W_dt, b_dt, DR, T, D, R, 96, R);

  // 5) chunked scan + gate -> Y
  scan_gate_kernel<<<dim3(D / 256, L / CH, B), 256, 0, stream>>>(
      DR, U, XP, XR, A_log, Y, L, D, CH, 2 * D);

  // 6) out_proj: out = Y @ W_out^T + b_out      M=2048 N=1024 K=2048
  gemm_f32_wmma<4><<<dim3(DM / 64, T / 64), gblk, 0, stream>>>(
      Y, W_out, b_out, out, T, DM, D, D, D);
}